// FilterDetection_9457517986253
// MI455X (gfx1250) — compile-verified
//
#include <hip/hip_runtime.h>
#include <hip/hip_bf16.h>
#include <cstdint>

// ---------------------------------------------------------------------------
// FilterDetection for MI455X (gfx1250, wave32).
// B=1, N=200000, C=81, K=5000 (PERF_COUNT), P=100 (PROPOSAL_COUNT)
// ---------------------------------------------------------------------------

#define NN       200000
#define NC       81
#define KTOP     5000
#define NPROP    100
#define NCLS     80          // classes 1..80 participate in NMS
#define NEGV     (-1.0e9f)
#define NEGH     (-5.0e8f)   // NEG/2 threshold
#define MAXR     4.135166556742356f   // |log(16/1000)|
#define NSORT    8192

// monotonic float -> uint key (bigger key == bigger float)
__device__ __forceinline__ unsigned fkey(float x) {
  unsigned u = __float_as_uint(x);
  return (u & 0x80000000u) ? ~u : (u | 0x80000000u);
}

// wave32 argmax reduce (ties -> lower index), lane 0 holds result
__device__ __forceinline__ void wave_argmax(float& v, int& i) {
  #pragma unroll
  for (int d = 16; d > 0; d >>= 1) {
    float v2 = __shfl_down(v, d, 32);
    int   i2 = __shfl_down(i, d, 32);
    if (v2 > v || (v2 == v && i2 < i)) { v = v2; i = i2; }
  }
}

// ---------------------------------------------------------------------------
// K1: per-anchor class max / argmax + validity.  Logits tile is staged into
// LDS with the CDNA5 async-to-LDS path (ASYNCcnt): b128 per lane (16 B),
// fully coalesced 65 MB streaming at HBM rate while the wave keeps issuing.
// ---------------------------------------------------------------------------
#define TILE_R 128
__global__ __launch_bounds__(TILE_R)
void k1_score(const float* __restrict__ score,
              const float* __restrict__ logits,
              const float* __restrict__ anchors,
              float* __restrict__ masked, int n) {
  __shared__ float tile[TILE_R * NC];          // 41472 B
  const int tid  = threadIdx.x;
  const int base = blockIdx.x * TILE_R;
  const int rows = min(TILE_R, n - base);
  const int words = rows * NC;
  const int v4 = words >> 2;                   // 16-byte chunks

  unsigned long long src = (unsigned long long)(const void*)(logits + (size_t)base * NC);
  unsigned lds_base = (unsigned)(uintptr_t)(&tile[0]);   // low 32 bits = LDS offset

  // bulk: 16 B per lane per issue (GVS mode: mem = SGPR64 + VGPR32 offset)
  for (int j = tid; j < v4; j += TILE_R) {
    unsigned off = (unsigned)(j * 16);
    unsigned lds_addr = lds_base + off;
    asm volatile("global_load_async_to_lds_b128 %0, %1, %2"
                 :: "v"(lds_addr), "v"(off), "s"(src) : "memory");
  }
  // remainder dwords (words % 4) — zero for full tiles
  for (int j = (v4 << 2) + tid; j < words; j += TILE_R) {
    unsigned off = (unsigned)(j * 4);
    unsigned lds_addr = lds_base + off;
    asm volatile("global_load_async_to_lds_b32 %0, %1, %2"
                 :: "v"(lds_addr), "v"(off), "s"(src) : "memory");
  }
  asm volatile("s_wait_asynccnt 0" ::: "memory");
  __syncthreads();

  if (tid < rows) {
    const int i = base + tid;
    const float s = score[i];
    float m = -3.0e38f; int lbl = 0;
    #pragma unroll 3
    for (int c = 0; c < NC; ++c) {
      float v = s * tile[tid * NC + c];
      if (v > m) { m = v; lbl = c; }
    }
    float4 a = *(const float4*)(anchors + (size_t)i * 4);
    bool valid = (a.z <= 1.0f) && (a.w <= 1.0f) && (a.x >= 0.0f) && (a.y >= 0.0f);
    valid = valid && (m >= 0.05f) && (lbl > 0);
    masked[i] = valid ? m : NEGV;
  }
}

// ---------------------------------------------------------------------------
// K2: exact top-K via 2-level (12+12 bit) radix select + bitonic sort.
// ---------------------------------------------------------------------------
__global__ void k2_hist1(const float* __restrict__ masked, int n, int* __restrict__ hist) {
  __shared__ int h[4096];
  for (int b = threadIdx.x; b < 4096; b += blockDim.x) h[b] = 0;
  __syncthreads();
  const int stride = gridDim.x * blockDim.x;
  for (int i = blockIdx.x * blockDim.x + threadIdx.x; i < n; i += stride)
    atomicAdd(&h[fkey(masked[i]) >> 20], 1);
  __syncthreads();
  for (int b = threadIdx.x; b < 4096; b += blockDim.x)
    if (h[b]) atomicAdd(&hist[b], h[b]);
}

// scal[0]=T1, scal[1]=G1, scal[2]=thr24, scal[3]=compaction counter
__global__ void k2_find1(const int* __restrict__ hist1, int* __restrict__ scal, int K) {
  int acc = 0;
  for (int b = 4095; b >= 0; --b) {
    int c = hist1[b];
    if (acc + c >= K) { scal[0] = b; scal[1] = acc; return; }
    acc += c;
  }
  scal[0] = 0; scal[1] = acc;
}

__global__ void k2_hist2(const float* __restrict__ masked, int n,
                         const int* __restrict__ scal, int* __restrict__ hist2) {
  const int T1 = scal[0];
  const int stride = gridDim.x * blockDim.x;
  for (int i = blockIdx.x * blockDim.x + threadIdx.x; i < n; i += stride) {
    unsigned k = fkey(masked[i]);
    if ((int)(k >> 20) == T1) atomicAdd(&hist2[(k >> 8) & 0xFFFu], 1);
  }
}

__global__ void k2_find2(const int* __restrict__ hist2, int* __restrict__ scal, int K) {
  const int T1 = scal[0];
  int acc = scal[1];
  for (int b = 4095; b >= 0; --b) {
    int c = hist2[b];
    if (acc + c >= K) { scal[2] = (T1 << 12) | b; return; }
    acc += c;
  }
  scal[2] = (T1 << 12);
}

__global__ void k2_compact(const float* __restrict__ masked, int n,
                           int* __restrict__ scal,
                           unsigned long long* __restrict__ sortbuf) {
  const unsigned thr = (unsigned)scal[2];
  const int stride = gridDim.x * blockDim.x;
  for (int i = blockIdx.x * blockDim.x + threadIdx.x; i < n; i += stride) {
    unsigned k = fkey(masked[i]);
    if ((k >> 8) >= thr) {
      int pos = atomicAdd(&scal[3], 1);
      if (pos < NSORT)
        sortbuf[pos] = ((unsigned long long)(~k) << 32) | (unsigned)i;  // asc sort = desc value, asc idx
    }
  }
}

// single block, 1024 threads, dynamic LDS = NSORT*8 = 64 KB
__global__ __launch_bounds__(1024)
void k2_sort(const unsigned long long* __restrict__ sortbuf,
             const int* __restrict__ scal, int* __restrict__ top_idx, int K) {
  extern __shared__ unsigned long long A[];
  int M = scal[3]; if (M > NSORT) M = NSORT;
  for (int i = threadIdx.x; i < NSORT; i += blockDim.x)
    A[i] = (i < M) ? sortbuf[i] : ~0ull;
  __syncthreads();
  for (int k = 2; k <= NSORT; k <<= 1) {
    for (int j = k >> 1; j > 0; j >>= 1) {
      for (int i = threadIdx.x; i < NSORT; i += blockDim.x) {
        int ixj = i ^ j;
        if (ixj > i) {
          unsigned long long a = A[i], b = A[ixj];
          bool asc = ((i & k) == 0);
          if ((a > b) == asc) { A[i] = b; A[ixj] = a; }
        }
      }
      __syncthreads();
    }
  }
  for (int t = threadIdx.x; t < K; t += blockDim.x) {
    unsigned long long e = A[t];
    top_idx[t] = ((unsigned)(e >> 32) == 0xFFFFFFFFu) ? -1 : (int)(unsigned)e;
  }
}

// ---------------------------------------------------------------------------
// K3: gather logit rows, decode + clip boxes, areas.
// ---------------------------------------------------------------------------
__global__ __launch_bounds__(96)
void k3_gather(const float* __restrict__ score, const float* __restrict__ logits,
               const float* __restrict__ regress, const float* __restrict__ anchors,
               const int* __restrict__ top_idx,
               float* __restrict__ l_top, float* __restrict__ boxes,
               float* __restrict__ areas) {
  const int j = blockIdx.x;
  const int c = threadIdx.x;
  const int i = top_idx[j];
  if (i >= 0) {
    const float s = score[i];
    if (c < NC) l_top[(size_t)j * NC + c] = s * logits[(size_t)i * NC + c];
    if (c == 0) {
      float4 a = *(const float4*)(anchors + (size_t)i * 4);
      float4 r = *(const float4*)(regress + (size_t)i * 4);
      float wa = a.z - a.x, ha = a.w - a.y;
      float cxa = a.x + 0.5f * wa, cya = a.y + 0.5f * ha;
      float dw = fminf(fmaxf(r.z, -MAXR), MAXR);
      float dh = fminf(fmaxf(r.w, -MAXR), MAXR);
      float cx = cxa + r.x * wa, cy = cya + r.y * ha;
      float w = wa * expf(dw), h = ha * expf(dh);
      float x1 = fminf(fmaxf(cx - 0.5f * w, 0.0f), 1.0f);
      float y1 = fminf(fmaxf(cy - 0.5f * h, 0.0f), 1.0f);
      float x2 = fminf(fmaxf(cx + 0.5f * w, 0.0f), 1.0f);
      float y2 = fminf(fmaxf(cy + 0.5f * h, 0.0f), 1.0f);
      boxes[j * 4 + 0] = x1; boxes[j * 4 + 1] = y1;
      boxes[j * 4 + 2] = x2; boxes[j * 4 + 3] = y2;
      areas[j] = (x2 - x1) * (y2 - y1);
    }
  } else {
    if (c < NC) l_top[(size_t)j * NC + c] = NEGV;
    if (c == 0) {
      boxes[j * 4 + 0] = 0.f; boxes[j * 4 + 1] = 0.f;
      boxes[j * 4 + 2] = 0.f; boxes[j * 4 + 3] = 0.f;
      areas[j] = 0.f;
    }
  }
}

// ---------------------------------------------------------------------------
// K4: per-class hard NMS, fully LDS-resident (120 KB dynamic LDS / WG).
// 80 blocks x 1024 threads (32 wave32 waves); shfl-based argmax (2 barriers
// per round instead of 10).
// ---------------------------------------------------------------------------
__global__ __launch_bounds__(1024)
void k4_nms(const int* __restrict__ top_idx, const float* __restrict__ l_top,
            const float* __restrict__ boxes, const float* __restrict__ areas,
            int* __restrict__ cand_idx, float* __restrict__ cand_sc) {
  extern __shared__ float S[];         // sc | x1 | y1 | x2 | y2 | area  (6 * 5000)
  float* sc = S;
  float* X1 = S + KTOP;  float* Y1 = S + 2 * KTOP;
  float* X2 = S + 3 * KTOP; float* Y2 = S + 4 * KTOP; float* AR = S + 5 * KTOP;
  __shared__ float rv[32];
  __shared__ int   ri[32];
  __shared__ float s_bv;
  __shared__ int   s_bi;

  const int cls = blockIdx.x + 1;      // classes 1..80
  const int tid = threadIdx.x;

  for (int j = tid; j < KTOP; j += 1024) {
    int i = top_idx[j];
    sc[j] = (i >= 0) ? l_top[(size_t)j * NC + cls] : NEGV;
    float4 b = *(const float4*)(boxes + j * 4);
    X1[j] = b.x; Y1[j] = b.y; X2[j] = b.z; Y2[j] = b.w;
    AR[j] = areas[j];
  }
  __syncthreads();

  for (int p = 0; p < NPROP; ++p) {
    // block argmax (ties -> lowest index, matching jnp.argmax)
    float bv = -3.0e38f; int bi = 0;
    for (int j = tid; j < KTOP; j += 1024) {
      float v = sc[j];
      if (v > bv) { bv = v; bi = j; }
    }
    wave_argmax(bv, bi);
    if ((tid & 31) == 0) { rv[tid >> 5] = bv; ri[tid >> 5] = bi; }
    __syncthreads();
    if (tid < 32) {
      float v = rv[tid]; int i2 = ri[tid];
      wave_argmax(v, i2);
      if (tid == 0) {
        s_bv = v; s_bi = i2;
        if (v > NEGH) {
          cand_idx[(cls - 1) * NPROP + p] = i2;
          cand_sc [(cls - 1) * NPROP + p] = v;
        }
      }
    }
    __syncthreads();
    const float bestv = s_bv;
    const int best = s_bi;
    if (bestv <= NEGH) break;          // uniform: nothing valid remains

    const float bx1 = X1[best], by1 = Y1[best], bx2 = X2[best], by2 = Y2[best];
    const float ba = AR[best];
    for (int j = tid; j < KTOP; j += 1024) {
      float v = sc[j];
      if (v > NEGH) {
        float ix1 = fmaxf(bx1, X1[j]);
        float iy1 = fmaxf(by1, Y1[j]);
        float ix2 = fminf(bx2, X2[j]);
        float iy2 = fminf(by2, Y2[j]);
        float inter = fmaxf(ix2 - ix1, 0.0f) * fmaxf(iy2 - iy1, 0.0f);
        float iou = inter / fmaxf(AR[j] + ba - inter, 1e-12f);
        if (iou > 0.5f || j == best) sc[j] = NEGV;
      }
    }
    __syncthreads();
  }
}

// ---------------------------------------------------------------------------
// K5: sorted top-100 of the 8000 per-class candidates.
// ---------------------------------------------------------------------------
__global__ __launch_bounds__(1024)
void k5_top100(const int* __restrict__ cand_idx, const float* __restrict__ cand_sc,
               int* __restrict__ sel, float* __restrict__ val) {
  __shared__ float s[NCLS * NPROP];    // 8000 floats
  __shared__ float rv[32];
  __shared__ int   ri[32];
  const int tid = threadIdx.x;
  for (int t = tid; t < NCLS * NPROP; t += 1024)
    s[t] = (cand_idx[t] >= 0) ? cand_sc[t] : NEGV;
  __syncthreads();
  for (int p = 0; p < NPROP; ++p) {
    float bv = -3.0e38f; int bi = 0;
    for (int t = tid; t < NCLS * NPROP; t += 1024) {
      float v = s[t];
      if (v > bv) { bv = v; bi = t; }
    }
    wave_argmax(bv, bi);
    if ((tid & 31) == 0) { rv[tid >> 5] = bv; ri[tid >> 5] = bi; }
    __syncthreads();
    if (tid < 32) {
      float v = rv[tid]; int i2 = ri[tid];
      wave_argmax(v, i2);
      if (tid == 0) {
        val[p] = v;
        sel[p] = cand_idx[i2];
        s[i2] = -3.0e38f;
      }
    }
    __syncthreads();
  }
}

// ---------------------------------------------------------------------------
// K6: write outputs (out_logit 100x81, then out_box 100x4).
// ---------------------------------------------------------------------------
__global__ __launch_bounds__(128)
void k6_out(const int* __restrict__ sel, const float* __restrict__ val,
            const float* __restrict__ l_top, const float* __restrict__ boxes,
            float* __restrict__ out) {
  const int p = blockIdx.x;
  const int c = threadIdx.x;
  const bool ok = val[p] > NEGH;
  int sl = sel[p]; if (sl < 0) sl = 0;
  if (c < NC) out[(size_t)p * NC + c] = ok ? l_top[(size_t)sl * NC + c] : 0.0f;
  if (c >= 96 && c < 100) {
    int k = c - 96;
    out[NPROP * NC + p * 4 + k] = ok ? boxes[sl * 4 + k] : 0.0f;
  }
}

// ---------------------------------------------------------------------------
// Host launcher
// ---------------------------------------------------------------------------
extern "C" void kernel_launch(void* const* d_in, const int* in_sizes, int n_in,
                              void* d_out, int out_size, void* d_ws, size_t ws_size,
                              hipStream_t stream) {
  const float* score   = (const float*)d_in[0];   // (1,N,1)
  const float* logits  = (const float*)d_in[1];   // (1,N,81)
  const float* regress = (const float*)d_in[2];   // (1,N,4)
  const float* anchors = (const float*)d_in[3];   // (N,4)
  float* out = (float*)d_out;                     // 8100 + 400 floats
  const int n = in_sizes[0];                      // N = 200000

  char* ws = (char*)d_ws;
  // workspace layout (all 16B aligned)
  float*              masked   = (float*)(ws + 0);                  // N floats
  int*                hist1    = (int*)(ws + 800000);               // 4096
  int*                hist2    = (int*)(ws + 816384);               // 4096
  int*                scal     = (int*)(ws + 832768);               // scalars
  unsigned long long* sortbuf  = (unsigned long long*)(ws + 833024);// 8192 u64
  int*                top_idx  = (int*)(ws + 898560);               // 5000
  float*              l_top    = (float*)(ws + 918560);             // 5000*81
  float*              boxes    = (float*)(ws + 2538560);            // 5000*4
  float*              areas    = (float*)(ws + 2618560);            // 5000
  int*                cand_idx = (int*)(ws + 2638560);              // 8000
  float*              cand_sc  = (float*)(ws + 2670560);            // 8000
  int*                sel100   = (int*)(ws + 2702560);              // 100
  float*              val100   = (float*)(ws + 2702960);            // 100

  // dynamic-LDS opt-ins (WGP has 320 KB)
  hipFuncSetAttribute((const void*)k2_sort, hipFuncAttributeMaxDynamicSharedMemorySize,
                      NSORT * 8);
  hipFuncSetAttribute((const void*)k4_nms, hipFuncAttributeMaxDynamicSharedMemorySize,
                      6 * KTOP * 4);

  // zero hist1 + hist2 + scalars (contiguous), poison cand_idx with -1
  hipMemsetAsync(hist1, 0, 16384 + 16384 + 256, stream);
  hipMemsetAsync(cand_idx, 0xFF, NCLS * NPROP * 4, stream);

  // 1) per-anchor score/validity (async-LDS staged, b128)
  k1_score<<<(n + TILE_R - 1) / TILE_R, TILE_R, 0, stream>>>(score, logits, anchors, masked, n);

  // 2) exact top-K
  k2_hist1<<<512, 256, 0, stream>>>(masked, n, hist1);
  k2_find1<<<1, 1, 0, stream>>>(hist1, scal, KTOP);
  k2_hist2<<<512, 256, 0, stream>>>(masked, n, scal, hist2);
  k2_find2<<<1, 1, 0, stream>>>(hist2, scal, KTOP);
  k2_compact<<<512, 256, 0, stream>>>(masked, n, scal, sortbuf);
  k2_sort<<<1, 1024, NSORT * 8, stream>>>(sortbuf, scal, top_idx, KTOP);

  // 3) gather + decode
  k3_gather<<<KTOP, 96, 0, stream>>>(score, logits, regress, anchors, top_idx,
                                     l_top, boxes, areas);

  // 4) per-class NMS (80 classes, LDS-resident)
  k4_nms<<<NCLS, 1024, 6 * KTOP * 4, stream>>>(top_idx, l_top, boxes, areas,
                                               cand_idx, cand_sc);

  // 5) final sorted top-100
  k5_top100<<<1, 1024, 0, stream>>>(cand_idx, cand_sc, sel100, val100);

  // 6) outputs
  k6_out<<<NPROP, 128, 0, stream>>>(sel100, val100, l_top, boxes, out);

  (void)n_in; (void)out_size; (void)ws_size;
}